// DecoderLayerStep_36867999269468
// MI455X (gfx1250) — compile-verified
//
#include <hip/hip_runtime.h>
#include <math.h>

// Problem constants (from reference): B=32, H=16, D=1024, DK=DV=64, DI=4096,
// T_self = 2047+1 = 2048, S_enc = 1024, scale = sqrt(64) = 8.
#define Bsz   32
#define Hn    16
#define Dm    1024
#define DKv   64
#define DIm   4096
#define TSELF 2048
#define TPREV 2047
#define SENC  1024

typedef __attribute__((ext_vector_type(2))) float v2f;
typedef __attribute__((ext_vector_type(4))) float f4;
typedef __attribute__((ext_vector_type(8))) float v8f;

__device__ __forceinline__ float wred_max(float v) {
#pragma unroll
  for (int i = 16; i > 0; i >>= 1) v = fmaxf(v, __shfl_xor(v, i, 32));
  return v;
}
__device__ __forceinline__ float wred_sum(float v) {
#pragma unroll
  for (int i = 16; i > 0; i >>= 1) v += __shfl_xor(v, i, 32);
  return v;
}

// ---------------------------------------------------------------------------
// WMMA f32 GEMM: C[32 x N] = A[32 x K] * B[K x N] (+bias, optional ReLU).
// 8 waves per block; wave -> one 16x16 C tile (2 M-tiles x 4 N-tiles = 32x64
// per block). blockIdx.y batches over heads (B/C strided) for the per-head
// q/k/v projections where W is [H][D][64].
// Fragment layouts per CDNA5 ISA 7.12.2:
//   A 16x4 f32 : lane<16 -> M=lane,    VGPR0=K0, VGPR1=K1
//                lane>=16 -> M=lane-16, VGPR0=K2, VGPR1=K3
//   B 4x16 f32 : lane<16 -> N=lane, VGPR0=K0, VGPR1=K1 ; lane>=16 -> K2/K3
//   C/D 16x16  : VGPR j -> M = j + 8*(lane>=16), N = lane&15
// ---------------------------------------------------------------------------
__global__ __launch_bounds__(256) void gemm32_wmma(
    const float* __restrict__ A, int lda,
    const float* __restrict__ Bm, int ldb, long sB,
    const float* __restrict__ bias,
    float* __restrict__ C, int ldc, long sC,
    int K, int relu)
{
  const float* Bp = Bm + (long)blockIdx.y * sB;
  float*       Cp = C  + (long)blockIdx.y * sC;
  const int wave = threadIdx.x >> 5;
  const int lane = threadIdx.x & 31;
  const int mt   = wave & 1;         // M tile: 0..1
  const int nt   = wave >> 1;        // N tile: 0..3
  const int row  = lane & 15;
  const int khi  = (lane >> 4) << 1; // 0 or 2 (per-lane K offset)
  const int m    = mt * 16 + row;
  const int ncol = blockIdx.x * 64 + nt * 16 + row; // this lane's column

  const float* Arow = A  + (long)m * lda + khi;
  const float* Bcol = Bp + (long)khi * ldb + ncol;

  v8f acc = {};
#pragma unroll 4
  for (int kk = 0; kk < K; kk += 4) {
    float2 av = *(const float2*)(Arow + kk);     // A[m][kk+khi], A[m][kk+khi+1]
    v2f a; a.x = av.x; a.y = av.y;
    v2f b;
    b.x = Bcol[(long)kk * ldb];                  // B[kk+khi  ][ncol]
    b.y = Bcol[(long)kk * ldb + ldb];            // B[kk+khi+1][ncol]
    acc = __builtin_amdgcn_wmma_f32_16x16x4_f32(
        /*neg_a=*/false, a, /*neg_b=*/false, b,
        /*c_mod=*/(short)0, acc, /*reuse_a=*/false, /*reuse_b=*/false);
  }

  const int chalf = (lane >> 4) * 8;
#pragma unroll
  for (int j = 0; j < 8; j++) {
    int crow = mt * 16 + chalf + j;
    float v = acc[j];
    if (bias) v += bias[ncol];
    if (relu) v = fmaxf(v, 0.0f);
    Cp[(long)crow * ldc + ncol] = v;
  }
}

// ---------------------------------------------------------------------------
// Single-query attention (flash-decode), one workgroup per (b,h).
//   scores -> softmax -> attn weights out (required output) -> ctx = P*V.
// K/V streamed once with non-temporal b128 loads (860 MB >> 192 MB L2, so NT
// avoids thrashing). Row T-1 of self-attn K/V comes from k_new/v_new.
// ---------------------------------------------------------------------------
__global__ __launch_bounds__(256) void attn_decode(
    const float* __restrict__ q,              // [B,H,64]
    const float* __restrict__ Kc,             // [B,H,Tc,64]
    const float* __restrict__ Vc,             // [B,H,Tc,64]
    const float* __restrict__ knew,           // [B,H,64] or null
    const float* __restrict__ vnew,           // [B,H,64] or null
    const unsigned char* __restrict__ mask,   // [B,T] (jnp bool_, 1 byte)
    float* __restrict__ attn,                 // [B,H,T]
    float* __restrict__ ctx,                  // [B,H,64]
    int T, int Tc)
{
  __shared__ float sc[TSELF];
  __shared__ float qs[64];
  __shared__ float red[8];

  const int bh   = blockIdx.x;
  const int b    = bh >> 4;            // H == 16
  const int tid  = threadIdx.x;
  const int lane = tid & 31, wid = tid >> 5;

  if (tid < 64) qs[tid] = q[(long)bh * 64 + tid];
  __syncthreads();

  const f4* q4 = (const f4*)qs;
  const unsigned char* mrow = mask + (long)b * T;

  // ---- scores + running max (each thread owns rows t = tid + k*256) ----
  float lmax = -INFINITY;
  for (int t = tid; t < T; t += 256) {
    const float* krow = (t < Tc) ? (Kc + ((long)bh * Tc + t) * 64)
                                 : (knew + (long)bh * 64);
    const f4* k4 = (const f4*)krow;
    float s = 0.0f;
#pragma unroll
    for (int i = 0; i < 16; i++) {
      f4 kv = __builtin_nontemporal_load(&k4[i]);
      f4 qv = q4[i];
      s += kv.x * qv.x + kv.y * qv.y + kv.z * qv.z + kv.w * qv.w;
    }
    s *= 0.125f;                         // 1/sqrt(64)
    if (mrow[t]) s = -INFINITY;
    sc[t] = s;
    lmax = fmaxf(lmax, s);
  }
  lmax = wred_max(lmax);
  if (lane == 0) red[wid] = lmax;
  __syncthreads();
  float gmax = red[0];
#pragma unroll
  for (int i = 1; i < 8; i++) gmax = fmaxf(gmax, red[i]);
  __syncthreads();

  // ---- exp + sum ----
  float lsum = 0.0f;
  for (int t = tid; t < T; t += 256) {
    float p = __expf(sc[t] - gmax);
    sc[t] = p;
    lsum += p;
  }
  lsum = wred_sum(lsum);
  if (lane == 0) red[wid] = lsum;
  __syncthreads();
  float gsum = 0.0f;
#pragma unroll
  for (int i = 0; i < 8; i++) gsum += red[i];
  const float inv = 1.0f / gsum;

  // ---- normalized attention weights out: b128 stores, coalesced ----
  for (int t4 = tid * 4; t4 < T; t4 += 1024) {
    f4 p = ((const f4*)sc)[t4 >> 2];
    p *= inv;
    *(f4*)(attn + (long)bh * T + t4) = p;
  }

  // ---- ctx[v] = inv * sum_t p[t] * V[t][v] ----
  // tid -> (f4 column group = tid&15, t-chunk = tid>>4). 16 lanes x 16B cover
  // one 256 B V row exactly once, as full-width b128 NT loads.
  const int vv4 = (tid & 15) * 4;
  const int ch  = tid >> 4;            // 0..15
  f4 acc = {0.0f, 0.0f, 0.0f, 0.0f};
  for (int t = ch; t < T; t += 16) {
    f4 vval = (t < Tc)
        ? __builtin_nontemporal_load((const f4*)(Vc + ((long)bh * Tc + t) * 64 + vv4))
        : *(const f4*)(vnew + (long)bh * 64 + vv4);
    acc += sc[t] * vval;
  }
  __syncthreads();                     // all sc reads done before staging reuse
  ((f4*)sc)[ch * 16 + (tid & 15)] = acc;   // sc[ch*64 + vv4 ..]
  __syncthreads();
  if (tid < 64) {
    float r = 0.0f;
#pragma unroll
    for (int c = 0; c < 16; c++) r += sc[c * 64 + tid];
    ctx[(long)bh * 64 + tid] = r * inv;
  }
}

// ---------------------------------------------------------------------------
// out[b,:] = LayerNorm(y[b,:] + r[b,:]) * g + beta   (D = 1024, one WG/row)
// ---------------------------------------------------------------------------
__global__ __launch_bounds__(256) void add_ln(
    const float* __restrict__ y, const float* __restrict__ r,
    const float* __restrict__ g, const float* __restrict__ be,
    float* __restrict__ out)
{
  __shared__ float reds[8], reds2[8];
  const int b = blockIdx.x, tid = threadIdx.x;
  const int lane = tid & 31, wid = tid >> 5;
  const int idx4 = tid * 4;            // 256 threads x f4 = 1024

  f4 t = *(const f4*)(y + (long)b * Dm + idx4);
  f4 rr = *(const f4*)(r + (long)b * Dm + idx4);
  t += rr;
  float s  = t.x + t.y + t.z + t.w;
  float s2 = t.x * t.x + t.y * t.y + t.z * t.z + t.w * t.w;

  s = wred_sum(s); s2 = wred_sum(s2);
  if (lane == 0) { reds[wid] = s; reds2[wid] = s2; }
  __syncthreads();
  float ts = 0.0f, ts2 = 0.0f;
#pragma unroll
  for (int i = 0; i < 8; i++) { ts += reds[i]; ts2 += reds2[i]; }
  const float mean = ts * (1.0f / Dm);
  const float var  = ts2 * (1.0f / Dm) - mean * mean;
  const float inv  = rsqrtf(var + 1e-6f);

  f4 gg = *(const f4*)(g + idx4);
  f4 bb = *(const f4*)(be + idx4);
  f4 o;
  o.x = (t.x - mean) * inv * gg.x + bb.x;
  o.y = (t.y - mean) * inv * gg.y + bb.y;
  o.z = (t.z - mean) * inv * gg.z + bb.z;
  o.w = (t.w - mean) * inv * gg.w + bb.w;
  *(f4*)(out + (long)b * Dm + idx4) = o;
}

// ---------------------------------------------------------------------------
extern "C" void kernel_launch(void* const* d_in, const int* in_sizes, int n_in,
                              void* d_out, int out_size, void* d_ws, size_t ws_size,
                              hipStream_t stream)
{
  (void)in_sizes; (void)n_in; (void)out_size; (void)ws_size;

  const float* dec       = (const float*)d_in[0];
  const float* cache_k   = (const float*)d_in[1];
  const float* cache_v   = (const float*)d_in[2];
  const float* enc_k     = (const float*)d_in[3];
  const float* enc_v     = (const float*)d_in[4];
  const unsigned char* slf_mask = (const unsigned char*)d_in[5];
  const unsigned char* enc_mask = (const unsigned char*)d_in[6];
  const float* w_qs_s    = (const float*)d_in[7];
  const float* w_ks_s    = (const float*)d_in[8];
  const float* w_vs_s    = (const float*)d_in[9];
  const float* proj_w_s  = (const float*)d_in[10];
  const float* proj_b_s  = (const float*)d_in[11];
  const float* ln_g_s    = (const float*)d_in[12];
  const float* ln_b_s    = (const float*)d_in[13];
  const float* w_qs_e    = (const float*)d_in[14];
  const float* proj_w_e  = (const float*)d_in[15];
  const float* proj_b_e  = (const float*)d_in[16];
  const float* ln_g_e    = (const float*)d_in[17];
  const float* ln_b_e    = (const float*)d_in[18];
  const float* ffn_w1    = (const float*)d_in[19];
  const float* ffn_b1    = (const float*)d_in[20];
  const float* ffn_w2    = (const float*)d_in[21];
  const float* ffn_b2    = (const float*)d_in[22];
  const float* ln_g_f    = (const float*)d_in[23];
  const float* ln_b_f    = (const float*)d_in[24];

  float* out      = (float*)d_out;                       // [B,D]
  float* slf_attn = out + (long)Bsz * Dm;                // [B,H,2048]
  float* enc_attn = slf_attn + (long)Bsz * Hn * TSELF;   // [B,H,1024]

  float* ws    = (float*)d_ws;
  float* q_s   = ws;               // [B,H,64] = 32768
  float* k_new = q_s   + 32768;
  float* v_new = k_new + 32768;
  float* q_e   = v_new + 32768;
  float* ctx_s = q_e   + 32768;
  float* ctx_e = ctx_s + 32768;
  float* ybuf  = ctx_e + 32768;
  float* x1    = ybuf  + 32768;
  float* x2    = x1    + 32768;
  float* hbuf  = x2    + 32768;    // [B,DI] = 131072

  const dim3 blk(256);
  const long sW = (long)Dm * DKv;  // per-head weight stride

  // self-attn projections (per-head batched GEMMs)
  gemm32_wmma<<<dim3(1, Hn), blk, 0, stream>>>(dec, Dm, w_qs_s, DKv, sW, nullptr, q_s,   Hn*DKv, DKv, Dm, 0);
  gemm32_wmma<<<dim3(1, Hn), blk, 0, stream>>>(dec, Dm, w_ks_s, DKv, sW, nullptr, k_new, Hn*DKv, DKv, Dm, 0);
  gemm32_wmma<<<dim3(1, Hn), blk, 0, stream>>>(dec, Dm, w_vs_s, DKv, sW, nullptr, v_new, Hn*DKv, DKv, Dm, 0);

  // self attention over cache + new token
  attn_decode<<<Bsz * Hn, blk, 0, stream>>>(q_s, cache_k, cache_v, k_new, v_new,
                                            slf_mask, slf_attn, ctx_s, TSELF, TPREV);
  gemm32_wmma<<<dim3(Dm / 64, 1), blk, 0, stream>>>(ctx_s, Hn*DKv, proj_w_s, Dm, 0, proj_b_s, ybuf, Dm, 0, Hn*DKv, 0);
  add_ln<<<Bsz, blk, 0, stream>>>(ybuf, dec, ln_g_s, ln_b_s, x1);

  // cross attention against encoder K/V
  gemm32_wmma<<<dim3(1, Hn), blk, 0, stream>>>(x1, Dm, w_qs_e, DKv, sW, nullptr, q_e, Hn*DKv, DKv, Dm, 0);
  attn_decode<<<Bsz * Hn, blk, 0, stream>>>(q_e, enc_k, enc_v, nullptr, nullptr,
                                            enc_mask, enc_attn, ctx_e, SENC, SENC);
  gemm32_wmma<<<dim3(Dm / 64, 1), blk, 0, stream>>>(ctx_e, Hn*DKv, proj_w_e, Dm, 0, proj_b_e, ybuf, Dm, 0, Hn*DKv, 0);
  add_ln<<<Bsz, blk, 0, stream>>>(ybuf, x1, ln_g_e, ln_b_e, x2);

  // FFN
  gemm32_wmma<<<dim3(DIm / 64, 1), blk, 0, stream>>>(x2,   Dm,  ffn_w1, DIm, 0, ffn_b1, hbuf, DIm, 0, Dm,  1);
  gemm32_wmma<<<dim3(Dm  / 64, 1), blk, 0, stream>>>(hbuf, DIm, ffn_w2, Dm,  0, ffn_b2, ybuf, Dm,  0, DIm, 0);
  add_ln<<<Bsz, blk, 0, stream>>>(ybuf, x2, ln_g_f, ln_b_f, out);
}